// GraphNet_22686017257664
// MI455X (gfx1250) — compile-verified
//
#include <hip/hip_runtime.h>
#include <math.h>

typedef float v2f __attribute__((ext_vector_type(2)));
typedef float v8f __attribute__((ext_vector_type(8)));

#define NFEAT 256

// deg starts at 1.0 = self-loop weight contribution
__global__ void init_deg_kernel(float* __restrict__ deg, int N) {
    int i = blockIdx.x * blockDim.x + threadIdx.x;
    if (i < N) deg[i] = 1.0f;
}

// deg[col[e]] += attrs[e]
__global__ void deg_edges_kernel(const long long* __restrict__ col,
                                 const float* __restrict__ attrs,
                                 float* __restrict__ deg, int E) {
    int e = blockIdx.x * blockDim.x + threadIdx.x;
    if (e < E) atomicAdd(&deg[(int)col[e]], attrs[e]);
}

// h = x @ W via V_WMMA_F32_16X16X4_F32.
// One wave computes 16 rows; W[k] is broadcast into every B column so each
// column of the 16x16 accumulator holds the GEMV result.
__global__ void gemv_wmma_kernel(const float* __restrict__ x,
                                 const float* __restrict__ W,
                                 float* __restrict__ h, int N) {
    int wave = (blockIdx.x * blockDim.x + threadIdx.x) >> 5;
    int lane = threadIdx.x & 31;
    int row0 = wave * 16;
    if (row0 >= N) return;                 // uniform per wave (N % 16 == 0)

    int m    = row0 + (lane & 15);         // A: lanes 0-15 and 16-31 both hold M=0..15
    int ksub = (lane >> 4) << 1;           // lanes 0-15 -> K+0/K+1, lanes 16-31 -> K+2/K+3
    const float* __restrict__ xrow = x + (size_t)m * NFEAT;

    v8f c = {};
    for (int k = 0; k < NFEAT; k += 4) {
        v2f a, bm;
        a[0]  = xrow[k + ksub];
        a[1]  = xrow[k + ksub + 1];
        bm[0] = W[k + ksub];               // broadcast W across all 16 B columns
        bm[1] = W[k + ksub + 1];
        c = __builtin_amdgcn_wmma_f32_16x16x4_f32(
                /*neg_a=*/false, a, /*neg_b=*/false, bm,
                /*c_mod=*/(short)0, c, /*reuse_a=*/false, /*reuse_b=*/false);
    }
    // D layout: lane 0 has M=0..7 (N=0), lane 16 has M=8..15 (N=0)
    if (lane == 0) {
        #pragma unroll
        for (int v = 0; v < 8; ++v) h[row0 + v] = c[v];
    } else if (lane == 16) {
        #pragma unroll
        for (int v = 0; v < 8; ++v) h[row0 + 8 + v] = c[v];
    }
}

// dis = rsqrt(deg) (in place); acc seeded with self-loop message h[i]/deg[i]
__global__ void dis_acc_kernel(float* __restrict__ deg,
                               const float* __restrict__ h,
                               float* __restrict__ acc, int N) {
    int i = blockIdx.x * blockDim.x + threadIdx.x;
    if (i < N) {
        float r = rsqrtf(deg[i]);          // deg >= 1 always (self loop)
        deg[i] = r;
        acc[i] = h[i] * r * r;
    }
}

// acc[col] += dis[row] * attr * dis[col] * h[row]
__global__ void msg_edges_kernel(const long long* __restrict__ row,
                                 const long long* __restrict__ col,
                                 const float* __restrict__ attrs,
                                 const float* __restrict__ dis,
                                 const float* __restrict__ h,
                                 float* __restrict__ acc, int E) {
    int e = blockIdx.x * blockDim.x + threadIdx.x;
    if (e < E) {
        int r = (int)row[e];
        int t = (int)col[e];
        float nrm = dis[r] * attrs[e] * dis[t];
        atomicAdd(&acc[t], nrm * h[r]);
    }
}

// out = mish(acc + b)
__global__ void finish_kernel(const float* __restrict__ acc,
                              const float* __restrict__ b,
                              float* __restrict__ out, int N) {
    int i = blockIdx.x * blockDim.x + threadIdx.x;
    if (i < N) {
        float o  = acc[i] + b[0];
        float sp = (o > 20.0f) ? o : log1pf(__expf(o));   // stable softplus
        out[i] = o * tanhf(sp);
    }
}

extern "C" void kernel_launch(void* const* d_in, const int* in_sizes, int n_in,
                              void* d_out, int out_size, void* d_ws, size_t ws_size,
                              hipStream_t stream) {
    const float*     x     = (const float*)d_in[0];
    const long long* ei    = (const long long*)d_in[1];   // int64 [2, E]
    const float*     attrs = (const float*)d_in[2];
    const float*     W     = (const float*)d_in[3];
    const float*     b     = (const float*)d_in[4];

    int N = in_sizes[0] / NFEAT;       // 200000
    int E = in_sizes[2];               // 12800000
    const long long* row = ei;         // sources
    const long long* col = ei + E;     // targets

    float* deg = (float*)d_ws;         // N floats, becomes dis in place
    float* h   = deg + N;              // N floats
    float* acc = h + N;                // N floats
    float* out = (float*)d_out;

    const int TB = 256;
    int nBlocksN = (N + TB - 1) / TB;
    int nBlocksE = (E + TB - 1) / TB;

    init_deg_kernel<<<nBlocksN, TB, 0, stream>>>(deg, N);
    deg_edges_kernel<<<nBlocksE, TB, 0, stream>>>(col, attrs, deg, E);

    int waves  = (N + 15) / 16;                       // one wave per 16 rows
    int gblocks = (waves * 32 + TB - 1) / TB;
    gemv_wmma_kernel<<<gblocks, TB, 0, stream>>>(x, W, h, N);

    dis_acc_kernel<<<nBlocksN, TB, 0, stream>>>(deg, h, acc, N);
    msg_edges_kernel<<<nBlocksE, TB, 0, stream>>>(row, col, attrs, deg, h, acc, E);
    finish_kernel<<<nBlocksN, TB, 0, stream>>>(acc, b, out, N);
}